// myGraphConvolution_88545045774865
// MI455X (gfx1250) — compile-verified
//
#include <hip/hip_runtime.h>
#include <hip/hip_bf16.h>

// N = 50000 nodes, F_IN = F_OUT = 256, K = 3 Chebyshev polys, E = 1.6M edges each.
// Pipeline per k: support_k = input @ weight[k]  (fp32 WMMA) -> SpMM scatter of
// support_k into out. support_k (51.2 MB) stays L2-resident (192 MB L2) between
// the two kernels since they run back-to-back on the same stream.

typedef __attribute__((ext_vector_type(2))) float v2f;
typedef __attribute__((ext_vector_type(8))) float v8f;

#define FDIM 256

// ---------------------------------------------------------------------------
// out[n][f] = bias[f]   (d_out is poisoned; atomics accumulate on top of bias)
// ---------------------------------------------------------------------------
__global__ void cheb_bias_init_kernel(float* __restrict__ out,
                                      const float* __restrict__ bias,
                                      int total) {
    int idx = blockIdx.x * blockDim.x + threadIdx.x;
    if (idx < total) out[idx] = bias[idx & (FDIM - 1)];
}

// ---------------------------------------------------------------------------
// S[m][n] = sum_k A[m][k] * W[k][n]   for one Chebyshev weight slice.
// A: [N,256] row-major, W: [256,256] row-major, S: [N,256] row-major.
// One wave computes a 16x64 strip: 4 f32 WMMA accumulators, A-frag reused 4x.
// N is a multiple of 16 (50000 = 3125*16), so no tail handling needed and
// EXEC stays all-ones for WMMA.
// ---------------------------------------------------------------------------
__global__ void cheb_gemm_wmma_kernel(const float* __restrict__ A,
                                      const float* __restrict__ W,
                                      float* __restrict__ S,
                                      int ntasks) {
    const int lane = threadIdx.x & 31;
    const int wave = threadIdx.x >> 5;
    const int task = blockIdx.x * 8 + wave;     // (N/16) * 4 tasks
    if (task >= ntasks) return;

    const int mt = task >> 2;                   // 16-row tile index
    const int nt = task & 3;                    // 64-col supertile index
    const int m0 = mt * 16;
    const int n0 = nt * 64;

    // A-fragment addressing (16x4 f32, ISA layout):
    //   v0: K = koff   (lanes 0-15 -> koff=0, lanes 16-31 -> koff=2)
    //   v1: K = koff+1
    const int arow = m0 + (lane & 15);
    const int koff = (lane >> 4) << 1;          // 0 or 2
    // B-fragment addressing (4x16 f32): N = lane&15, same K split as A.
    const int ncol = lane & 15;

    v8f c0 = {}, c1 = {}, c2 = {}, c3 = {};

    const float* arow_p = A + (size_t)arow * FDIM + koff;
    const float* wbase  = W + (size_t)koff * FDIM + n0 + ncol;

    #pragma unroll 4
    for (int kk = 0; kk < FDIM; kk += 4) {
        v2f a;
        const float* ap = arow_p + kk;
        a.x = ap[0];
        a.y = ap[1];

        const float* wp = wbase + (size_t)kk * FDIM;
        v2f b0, b1, b2, b3;
        b0.x = wp[0];        b0.y = wp[FDIM];
        b1.x = wp[16];       b1.y = wp[FDIM + 16];
        b2.x = wp[32];       b2.y = wp[FDIM + 32];
        b3.x = wp[48];       b3.y = wp[FDIM + 48];

        c0 = __builtin_amdgcn_wmma_f32_16x16x4_f32(false, a, false, b0, (short)0, c0, false, false);
        c1 = __builtin_amdgcn_wmma_f32_16x16x4_f32(false, a, false, b1, (short)0, c1, false, false);
        c2 = __builtin_amdgcn_wmma_f32_16x16x4_f32(false, a, false, b2, (short)0, c2, false, false);
        c3 = __builtin_amdgcn_wmma_f32_16x16x4_f32(false, a, false, b3, (short)0, c3, false, false);
    }

    // C/D layout: VGPR i -> row m0 + i + 8*(lane>=16), col n0 + (lane&15)
    const int rbase = m0 + ((lane >> 4) << 3);
    #pragma unroll
    for (int i = 0; i < 8; ++i) {
        float* sp = S + (size_t)(rbase + i) * FDIM + n0 + (lane & 15);
        sp[0]  = c0[i];
        sp[16] = c1[i];
        sp[32] = c2[i];
        sp[48] = c3[i];
    }
}

// ---------------------------------------------------------------------------
// Edge scatter: out[rows[e]][:] += vals[e] * S[cols[e]][:]
// One wave per edge; 32 lanes x 2 float4 = 256 features, coalesced loads,
// 8 global_atomic_add_f32 per lane.
// ---------------------------------------------------------------------------
__global__ void cheb_spmm_scatter_kernel(const float* __restrict__ vals,
                                         const int*   __restrict__ rows,
                                         const int*   __restrict__ cols,
                                         const float* __restrict__ S,
                                         float*       __restrict__ out,
                                         int nedges) {
    const int gwave = (blockIdx.x * blockDim.x + threadIdx.x) >> 5;
    const int lane  = threadIdx.x & 31;
    if (gwave >= nedges) return;

    const float v = vals[gwave];
    const int   r = rows[gwave];
    const int   c = cols[gwave];

    const float4* sp = (const float4*)(S + (size_t)c * FDIM);
    float4 x0 = sp[lane];
    float4 x1 = sp[lane + 32];

    float* op0 = out + (size_t)r * FDIM + 4 * lane;
    float* op1 = op0 + 128;
    atomicAdd(op0 + 0, v * x0.x);
    atomicAdd(op0 + 1, v * x0.y);
    atomicAdd(op0 + 2, v * x0.z);
    atomicAdd(op0 + 3, v * x0.w);
    atomicAdd(op1 + 0, v * x1.x);
    atomicAdd(op1 + 1, v * x1.y);
    atomicAdd(op1 + 2, v * x1.z);
    atomicAdd(op1 + 3, v * x1.w);
}

// ---------------------------------------------------------------------------
extern "C" void kernel_launch(void* const* d_in, const int* in_sizes, int n_in,
                              void* d_out, int out_size, void* d_ws, size_t ws_size,
                              hipStream_t stream) {
    const float* input  = (const float*)d_in[0];   // [N, 256]
    const float* weight = (const float*)d_in[1];   // [3, 256, 256]
    const float* bias   = (const float*)d_in[2];   // [256]
    const float* tvals  = (const float*)d_in[3];   // [3, E]
    const int*   trows  = (const int*)  d_in[4];   // [3, E]
    const int*   tcols  = (const int*)  d_in[5];   // [3, E]
    float* out = (float*)d_out;
    float* sup = (float*)d_ws;                     // one support slice: N*256 floats

    const int N = in_sizes[0] / FDIM;              // 50000
    const int E = in_sizes[3] / 3;                 // 1600000

    // bias broadcast (also clears the poisoned output)
    {
        int total = N * FDIM;
        cheb_bias_init_kernel<<<(total + 255) / 256, 256, 0, stream>>>(out, bias, total);
    }

    const int ntasks  = (N / 16) * 4;              // 16-row x 64-col wave strips
    const int gblocks = (ntasks + 7) / 8;          // 8 waves per 256-thread block
    const int sblocks = (E + 7) / 8;               // 1 wave per edge

    for (int k = 0; k < 3; ++k) {
        cheb_gemm_wmma_kernel<<<gblocks, 256, 0, stream>>>(
            input, weight + (size_t)k * FDIM * FDIM, sup, ntasks);
        cheb_spmm_scatter_kernel<<<sblocks, 256, 0, stream>>>(
            tvals + (size_t)k * E, trows + (size_t)k * E, tcols + (size_t)k * E,
            sup, out, E);
    }
}